// ResidualBlock_16286515987120
// MI455X (gfx1250) — compile-verified
//
#include <hip/hip_runtime.h>

typedef __attribute__((ext_vector_type(2))) float v2f;
typedef __attribute__((ext_vector_type(8))) float v8f;

#define BN_  2
#define CC   64
#define HH   96
#define WW   96
#define NPIX (HH*WW)        // 9216
#define NPOS (BN_*NPIX)     // 18432
#define IC_  8
#define OC_  8
#define IL_  8
#define OL_  8
#define KK_  9
#define JD   72
#define TILES_X 6           // 96/16
#define NTILES (BN_*HH*TILES_X) // 1152
#define EPS_SIM 1e-8f
#define EPS_BN  1e-5f

// ---------------------------------------------------------------------------
// Fused capsule-conv (WMMA f32 16x16x4) + 3-iter cosine routing.
// One block = 16 consecutive pixels of one image row. 256 threads = 8 waves.
// Wave w owns j = w*9 .. w*9+8 (j = i*9 + k, i=input capsule, k=3x3 tap).
// For each o-pair g (2 output capsules packed in WMMA N=16):
//   priors tile (16 pos x 16 (oo,m) x 72 j) lives in LDS (72KB), then routing.
// ---------------------------------------------------------------------------
template<bool APPLY_BN>
__global__ __launch_bounds__(256) void caps_conv_route(
    const float* __restrict__ src,      // (B,64,96,96)
    const float* __restrict__ w,        // (8,8,3,3,8,8)
    const float* __restrict__ bn_mean,  // (64)  (unused if !APPLY_BN)
    const float* __restrict__ bn_inv,   // (64)  rsqrt(var+eps)
    const float* __restrict__ bn_g,     // (64)
    const float* __restrict__ bn_b,     // (64)
    float* __restrict__ dst)            // (B,64,96,96) pre-BN capsule output
{
  __shared__ float sPri[JD * 16 * 16];  // [j][pos][oo*8+m]  73728 B
  __shared__ float sLog[32 * 73];       // [task][j] softmax logits (padded)

  const int tid  = threadIdx.x;
  const int wave = tid >> 5;
  const int lane = tid & 31;
  const int hi   = lane >> 4;   // selects K half within a WMMA (0:{K0,K1} 1:{K2,K3})
  const int lo   = lane & 15;   // A: M row (position); B/D: N column

  const int blk = blockIdx.x;
  const int img = blk / (HH * TILES_X);
  const int rem = blk % (HH * TILES_X);
  const int py  = rem / TILES_X;
  const int px0 = (rem % TILES_X) * 16;

  // ---- Phase 0: load this wave's A fragments (patches, zero-padded, BN'd) ----
  float A[9][2][2];   // [j_local][k-half][component]; l = h*4 + hi*2 + comp
  #pragma unroll
  for (int jl = 0; jl < 9; ++jl) {
    const int j  = wave * 9 + jl;
    const int i  = j / KK_;
    const int k  = j % KK_;
    const int dy = k / 3 - 1;
    const int dx = k % 3 - 1;
    const int yy = py + dy;
    const int xx = px0 + lo + dx;
    const bool valid = (yy >= 0) && (yy < HH) && (xx >= 0) && (xx < WW);
    #pragma unroll
    for (int h = 0; h < 2; ++h) {
      const int l0 = h * 4 + hi * 2;
      #pragma unroll
      for (int comp = 0; comp < 2; ++comp) {
        const int c = i * IL_ + l0 + comp;
        float v = 0.0f;
        if (valid) {
          v = src[((img * CC + c) * HH + yy) * WW + xx];
          if (APPLY_BN)
            v = bn_g[c] * (v - bn_mean[c]) * bn_inv[c] + bn_b[c];
        }
        A[jl][h][comp] = v;
      }
    }
  }

  // ---- Per o-pair: WMMA fill of priors, then routing ----
  for (int g = 0; g < 4; ++g) {
    {
      const int oo = lo >> 3;          // which of the 2 output capsules in N
      const int m  = lo & 7;           // ol index
      const int o  = g * 2 + oo;
      #pragma unroll
      for (int jl = 0; jl < 9; ++jl) {
        const int j = wave * 9 + jl;
        const int i = j / KK_;
        const int k = j % KK_;
        const float* wb = w + (size_t)(((o * IC_ + i) * KK_ + k) * IL_) * OL_;
        v8f d = {0.f,0.f,0.f,0.f,0.f,0.f,0.f,0.f};
        #pragma unroll
        for (int h = 0; h < 2; ++h) {
          const int l0 = h * 4 + hi * 2;
          v2f a; a.x = A[jl][h][0]; a.y = A[jl][h][1];
          v2f b; b.x = wb[(l0 + 0) * OL_ + m];
                 b.y = wb[(l0 + 1) * OL_ + m];
          d = __builtin_amdgcn_wmma_f32_16x16x4_f32(
                false, a, false, b, (short)0, d, false, false);
        }
        #pragma unroll
        for (int r = 0; r < 8; ++r) {
          const int pos = r + 8 * hi;           // D: M = r + 8*(lane/16)
          sPri[(j * 16 + pos) * 16 + lo] = d[r]; // N = lane%16
        }
      }
    }
    __syncthreads();

    // ---- Routing: 32 tasks (pos 0..15 x oo 0..1) x 8 lanes (m) ----
    {
      const int task = tid >> 3;     // 0..31
      const int mm   = tid & 7;
      const int pos  = task & 15;
      const int oo2  = task >> 4;
      const int pidx = pos * 16 + oo2 * 8 + mm;

      float out_m = 0.f;
      #pragma unroll 8
      for (int j = 0; j < JD; ++j) out_m += sPri[j * 256 + pidx];
      out_m *= (1.0f / 72.0f);

      for (int it = 0; it < 3; ++it) {
        float n2 = out_m * out_m;
        n2 += __shfl_xor(n2, 1, 8);
        n2 += __shfl_xor(n2, 2, 8);
        n2 += __shfl_xor(n2, 4, 8);

        float mx = -3.4e38f;
        for (int j = 0; j < JD; ++j) {
          const float pv = sPri[j * 256 + pidx];
          float d1 = pv * out_m;
          float s1 = pv * pv;
          d1 += __shfl_xor(d1, 1, 8);  s1 += __shfl_xor(s1, 1, 8);
          d1 += __shfl_xor(d1, 2, 8);  s1 += __shfl_xor(s1, 2, 8);
          d1 += __shfl_xor(d1, 4, 8);  s1 += __shfl_xor(s1, 4, 8);
          const float lg = d1 / fmaxf(s1 + n2 - d1, EPS_SIM);
          if (mm == 0) sLog[task * 73 + j] = lg;
          mx = fmaxf(mx, lg);
        }
        __syncthreads();

        float esum = 0.f, num = 0.f;
        for (int j = 0; j < JD; ++j) {
          const float e = __expf(sLog[task * 73 + j] - mx);
          esum += e;
          num  += e * sPri[j * 256 + pidx];
        }
        out_m = num / esum;
        __syncthreads();   // before next iter rewrites sLog
      }

      const int ch = (g * 2 + oo2) * OL_ + mm;
      dst[((img * CC + ch) * HH + py) * WW + (px0 + pos)] = out_m;
    }
    __syncthreads();       // before next g rewrites sPri
  }
}

// ---------------------------------------------------------------------------
// Per-channel batch stats: mean and rsqrt(var+eps) over (B,H,W).
// ---------------------------------------------------------------------------
__global__ __launch_bounds__(256) void bn_stats(
    const float* __restrict__ y, float* __restrict__ mean, float* __restrict__ inv)
{
  __shared__ float ss[256];
  __shared__ float sq[256];
  const int ch = blockIdx.x;
  float s = 0.f, q = 0.f;
  for (int e = threadIdx.x; e < NPOS; e += 256) {
    const int im = e / NPIX;
    const int px = e % NPIX;
    const float v = y[(im * CC + ch) * NPIX + px];
    s += v; q += v * v;
  }
  ss[threadIdx.x] = s; sq[threadIdx.x] = q;
  __syncthreads();
  for (int st = 128; st > 0; st >>= 1) {
    if (threadIdx.x < st) {
      ss[threadIdx.x] += ss[threadIdx.x + st];
      sq[threadIdx.x] += sq[threadIdx.x + st];
    }
    __syncthreads();
  }
  if (threadIdx.x == 0) {
    const float mu  = ss[0] / (float)NPOS;
    const float var = sq[0] / (float)NPOS - mu * mu;
    mean[ch] = mu;
    inv[ch]  = rsqrtf(var + EPS_BN);
  }
}

// ---------------------------------------------------------------------------
// out = x + gamma*(y2-mean)*inv + beta
// ---------------------------------------------------------------------------
__global__ void finalize_residual(
    const float* __restrict__ x, const float* __restrict__ y2,
    const float* __restrict__ mean, const float* __restrict__ inv,
    const float* __restrict__ g, const float* __restrict__ b,
    float* __restrict__ out)
{
  const int idx = blockIdx.x * blockDim.x + threadIdx.x;
  if (idx >= BN_ * CC * NPIX) return;
  const int ch = (idx / NPIX) % CC;
  out[idx] = x[idx] + g[ch] * (y2[idx] - mean[ch]) * inv[ch] + b[ch];
}

extern "C" void kernel_launch(void* const* d_in, const int* in_sizes, int n_in,
                              void* d_out, int out_size, void* d_ws, size_t ws_size,
                              hipStream_t stream) {
  (void)in_sizes; (void)n_in; (void)out_size; (void)ws_size;
  const float* x  = (const float*)d_in[0];
  const float* w1 = (const float*)d_in[1];
  const float* g1 = (const float*)d_in[2];
  const float* b1 = (const float*)d_in[3];
  const float* w2 = (const float*)d_in[4];
  const float* g2 = (const float*)d_in[5];
  const float* b2 = (const float*)d_in[6];
  float* out = (float*)d_out;

  float* ws = (float*)d_ws;
  float* y1   = ws;                        // 1179648 floats
  float* y2   = ws + (size_t)BN_*CC*NPIX;  // 1179648 floats
  float* mu1  = y2 + (size_t)BN_*CC*NPIX;
  float* inv1 = mu1 + CC;
  float* mu2  = inv1 + CC;
  float* inv2 = mu2 + CC;

  // Layer 1: capsule conv + routing on raw x
  caps_conv_route<false><<<NTILES, 256, 0, stream>>>(x,  w1, mu1, inv1, g1, b1, y1);
  bn_stats<<<CC, 256, 0, stream>>>(y1, mu1, inv1);
  // Layer 2: BN(layer1) applied inline at patch load
  caps_conv_route<true ><<<NTILES, 256, 0, stream>>>(y1, w2, mu1, inv1, g1, b1, y2);
  bn_stats<<<CC, 256, 0, stream>>>(y2, mu2, inv2);
  // out = x + BN2(y2)
  const int total = BN_ * CC * NPIX;
  finalize_residual<<<(total + 255) / 256, 256, 0, stream>>>(x, y2, mu2, inv2, g2, b2, out);
}